// PolyConvFrame_61357902790934
// MI455X (gfx1250) — compile-verified
//
#include <hip/hip_runtime.h>

#define D 32u
#define CHUNK 512u
#define BLOCK 256u
#define SPMM_GRID 448u

typedef unsigned int u32;
typedef unsigned long long u64;

// ---------------- CDNA5 Tensor Data Mover support ----------------
#if defined(__has_builtin)
#  if __has_builtin(__builtin_amdgcn_tensor_load_to_lds) && __has_builtin(__builtin_amdgcn_s_wait_tensorcnt)
#    define USE_TDM 1
#  endif
#endif
#ifndef USE_TDM
#  define USE_TDM 0
#endif

#if USE_TDM
typedef u32 tdm_v4u __attribute__((ext_vector_type(4)));
typedef int tdm_v4i __attribute__((ext_vector_type(4)));
typedef int tdm_v8i __attribute__((ext_vector_type(8)));

// Issue (do NOT wait) a 1-D TDM copy: tile_words dwords from gaddr -> LDS byte
// offset lds_byte_addr. oob_words = dwords remaining from tile start (OOB -> 0).
__device__ __forceinline__ void tdm_issue_to_lds(u32 lds_byte_addr, u64 gaddr,
                                                 u32 tile_words, u32 oob_words) {
  // D# group 0: count=1 | lds_addr | global_addr[56:0] | type=2
  tdm_v4u g0;
  g0.x = 1u;
  g0.y = lds_byte_addr;
  g0.z = (u32)(gaddr & 0xFFFFFFFFull);
  g0.w = ((u32)(gaddr >> 32) & 0x01FFFFFFu) | (2u << 30);
  // D# group 1: data_size=2 (4B), tensor_dim0=oob_words, tensor_dim1=1,
  //             tile_dim0=tile_words, tile_dim1=1, stride0=oob_words
  tdm_v8i g1;
  g1.s0 = (int)(2u << 16);
  g1.s1 = (int)((oob_words & 0xFFFFu) << 16);
  g1.s2 = (int)((oob_words >> 16) | (1u << 16));
  g1.s3 = (int)(tile_words << 16);
  g1.s4 = 1;
  g1.s5 = (int)oob_words;
  g1.s6 = 0;
  g1.s7 = 0;
  // Groups 2/3: higher dims unused
  tdm_v4i g2; g2.x = 1; g2.y = 1; g2.z = 0; g2.w = 0;
  tdm_v4i g3; g3.x = 0; g3.y = 0; g3.z = 0; g3.w = 0;
#if __clang_major__ >= 23
  tdm_v8i g4; g4.s0 = 0; g4.s1 = 0; g4.s2 = 0; g4.s3 = 0;
  g4.s4 = 0; g4.s5 = 0; g4.s6 = 0; g4.s7 = 0;
  __builtin_amdgcn_tensor_load_to_lds(g0, g1, g2, g3, g4, 0);
#else
  __builtin_amdgcn_tensor_load_to_lds(g0, g1, g2, g3, 0);
#endif
}
#endif

// ---------------- small utility kernels ----------------
__global__ void k_alphas(const float* __restrict__ pe, float* __restrict__ alphas, u32 depth) {
  u32 i = threadIdx.x;
  if (i < depth) alphas[i] = tanhf(pe[i]);
}

__global__ void k_zero(float* __restrict__ p, u32 n) {
  u32 i = blockIdx.x * blockDim.x + threadIdx.x;
  if (i < n) p[i] = 0.0f;
}

__global__ void k_degree(const int* __restrict__ row, float* __restrict__ deg, u32 nE) {
  u32 e = blockIdx.x * blockDim.x + threadIdx.x;
  if (e < nE) unsafeAtomicAdd(deg + row[e], 1.0f);
}

__global__ void k_dinv(float* __restrict__ deg, u32 n) {
  u32 i = blockIdx.x * blockDim.x + threadIdx.x;
  if (i < n) {
    float d = deg[i];
    if (d < 0.5f) d += 1.0f;
    deg[i] = (float)(1.0 / sqrt((double)d));  // accurate deg^-0.5
  }
}

// Pack {row, col, val} into 12-byte records so the SpMM streams one array.
__global__ void k_prep(const int* __restrict__ ei, const float* __restrict__ ea,
                       const float* __restrict__ dinv, u32* __restrict__ erec, u32 nE) {
  u32 e = blockIdx.x * blockDim.x + threadIdx.x;
  if (e < nE) {
    int r = ei[e];
    int c = ei[nE + e];
    float v = dinv[r] * ea[e] * dinv[c];
    erec[3u * e + 0u] = (u32)r;
    erec[3u * e + 1u] = (u32)c;
    erec[3u * e + 2u] = __float_as_uint(v);
  }
}

// level-0 copy, float4-vectorized (8 quads per node; rows are 128B-aligned)
__global__ void k_copy_x(const float4* __restrict__ x4, float4* __restrict__ out4,
                         u32 n, u32 stride4) {
  u32 i = blockIdx.x * blockDim.x + threadIdx.x;
  if (i < n * 8u) {
    u32 node = i >> 3, q = i & 7u;
    out4[(u64)node * stride4 + q] = x4[i];
  }
}

__global__ void k_zero_slice(float4* __restrict__ out4, u32 n, u32 stride4, u32 off4) {
  u32 i = blockIdx.x * blockDim.x + threadIdx.x;
  if (i < n * 8u) {
    u32 node = i >> 3, q = i & 7u;
    out4[(u64)node * stride4 + off4 + q] = make_float4(0.f, 0.f, 0.f, 0.f);
  }
}

// ---------------- SpMM: h_next[row] += (alpha*val) * h_prev[col] ----------------
// 8 lanes per edge, float4 per lane => one full 128B cacheline gather per edge.
// Edge records double-buffered in LDS via the Tensor Data Mover (TENSORcnt).
__global__ void k_spmm(const u32* __restrict__ erec, u32 nEdge,
                       const float* __restrict__ hprev, u32 strideP,
                       float* __restrict__ hnext, u32 strideN,
                       const float* __restrict__ alphaPtr) {
  __shared__ u32 sh[2][CHUNK * 3u];
  const u32 tid = threadIdx.x;
  const u32 grp = tid >> 3;             // edge slot within a 32-edge pass
  const u32 lane4 = (tid & 7u) << 2;    // feature quad
  const float alpha = alphaPtr[0];      // uniform; alpha folded into edge weight

  u64 base = (u64)blockIdx.x * CHUNK;
  const u64 step = (u64)gridDim.x * CHUNK;
  if (base >= nEdge) return;

#if USE_TDM
  u32 cur = 0u;
  if (tid < 32u) {  // prologue: DMA chunk 0 into buffer 0
    const u64 rem0 = (u64)nEdge - base;
    const u32 nIn0 = (rem0 < (u64)CHUNK) ? (u32)rem0 : CHUNK;
    const u64 rw0 = rem0 * 3ull;
    tdm_issue_to_lds((u32)(size_t)&sh[0][0], (u64)(size_t)erec + base * 12ull,
                     nIn0 * 3u, (rw0 > 0xFFFFFFFFull) ? 0xFFFFFFFFu : (u32)rw0);
  }
  for (; base < nEdge; base += step) {
    const u64 rem = (u64)nEdge - base;
    const u32 nIn = (rem < (u64)CHUNK) ? (u32)rem : CHUNK;
    if (tid < 32u) __builtin_amdgcn_s_wait_tensorcnt(0);  // buffer `cur` ready
    __syncthreads();                                      // publish to all waves
    const u64 nb = base + step;
    if (nb < nEdge && tid < 32u) {  // prefetch next chunk into the other buffer
      const u64 remN = (u64)nEdge - nb;
      const u32 nInN = (remN < (u64)CHUNK) ? (u32)remN : CHUNK;
      const u64 rwN = remN * 3ull;
      tdm_issue_to_lds((u32)(size_t)&sh[cur ^ 1u][0],
                       (u64)(size_t)erec + nb * 12ull, nInN * 3u,
                       (rwN > 0xFFFFFFFFull) ? 0xFFFFFFFFu : (u32)rwN);
    }
    const u32* s0 = &sh[cur][0];
    for (u32 s = grp; s < nIn; s += BLOCK / 8u) {
      const u32 r = s0[3u * s + 0u];
      const u32 c = s0[3u * s + 1u];
      const float v = alpha * __uint_as_float(s0[3u * s + 2u]);
      const float4 hv =
          *reinterpret_cast<const float4*>(hprev + (u64)c * strideP + lane4);
      float* dst = hnext + (u64)r * strideN + lane4;
      unsafeAtomicAdd(dst + 0, v * hv.x);
      unsafeAtomicAdd(dst + 1, v * hv.y);
      unsafeAtomicAdd(dst + 2, v * hv.z);
      unsafeAtomicAdd(dst + 3, v * hv.w);
    }
    __syncthreads();  // all reads of buffer `cur` done before it is reused
    cur ^= 1u;
  }
#else
  for (; base < nEdge; base += step) {
    const u64 rem = (u64)nEdge - base;
    const u32 nIn = (rem < (u64)CHUNK) ? (u32)rem : CHUNK;
    const u32 words = nIn * 3u;
    const u64 gw = base * 3ull;
    for (u32 w = tid; w < words; w += BLOCK) sh[0][w] = erec[gw + w];
    __syncthreads();
    for (u32 s = grp; s < nIn; s += BLOCK / 8u) {
      const u32 r = sh[0][3u * s + 0u];
      const u32 c = sh[0][3u * s + 1u];
      const float v = alpha * __uint_as_float(sh[0][3u * s + 2u]);
      const float4 hv =
          *reinterpret_cast<const float4*>(hprev + (u64)c * strideP + lane4);
      float* dst = hnext + (u64)r * strideN + lane4;
      unsafeAtomicAdd(dst + 0, v * hv.x);
      unsafeAtomicAdd(dst + 1, v * hv.y);
      unsafeAtomicAdd(dst + 2, v * hv.z);
      unsafeAtomicAdd(dst + 3, v * hv.w);
    }
    __syncthreads();
  }
#endif
}

// ---------------- host launcher ----------------
extern "C" void kernel_launch(void* const* d_in, const int* in_sizes, int n_in,
                              void* d_out, int out_size, void* d_ws, size_t ws_size,
                              hipStream_t stream) {
  const float* x  = (const float*)d_in[0];
  const int*   ei = (const int*)d_in[1];
  const float* ea = (const float*)d_in[2];
  const float* pe = (const float*)d_in[3];
  float* out = (float*)d_out;

  const u32 nNode = (u32)(in_sizes[0] / (int)D);
  const u32 nEdge = (u32)in_sizes[2];
  const u32 depth = (u32)in_sizes[3];
  const u32 stride = (depth + 1u) * D;  // per-node output stride in floats

  // workspace layout: alphas | deg/dinv | packed edge records
  char* ws = (char*)d_ws;
  float* alphas = (float*)ws;
  size_t off = 256;
  float* deg = (float*)(ws + off);
  off += (((size_t)nNode * 4 + 255) & ~(size_t)255);
  u32* erec = (u32*)(ws + off);

  const u32 gN  = (nNode + BLOCK - 1u) / BLOCK;
  const u32 gE  = (nEdge + BLOCK - 1u) / BLOCK;
  const u32 gV4 = (nNode * 8u + BLOCK - 1u) / BLOCK;
  const u32 gC  = (nEdge + CHUNK - 1u) / CHUNK;
  const u32 gSP = (gC < SPMM_GRID) ? gC : SPMM_GRID;  // multi-chunk blocks => pipelined DMA

  k_alphas<<<1, 32, 0, stream>>>(pe, alphas, depth);
  k_zero<<<gN, BLOCK, 0, stream>>>(deg, nNode);
  k_degree<<<gE, BLOCK, 0, stream>>>(ei, deg, nEdge);
  k_dinv<<<gN, BLOCK, 0, stream>>>(deg, nNode);
  k_prep<<<gE, BLOCK, 0, stream>>>(ei, ea, deg, erec, nEdge);
  k_copy_x<<<gV4, BLOCK, 0, stream>>>((const float4*)x, (float4*)out, nNode, stride / 4u);

  for (u32 L = 1; L <= depth; ++L) {
    const u32 offN = L * D;
    const float* hprev;
    u32 strideP;
    if (L == 1u) { hprev = x; strideP = D; }
    else         { hprev = out + (u64)(L - 1u) * D; strideP = stride; }
    k_zero_slice<<<gV4, BLOCK, 0, stream>>>((float4*)out, nNode, stride / 4u, offN / 4u);
    k_spmm<<<gSP, BLOCK, 0, stream>>>(erec, nEdge, hprev, strideP, out + offN, stride,
                                      alphas + (L - 1u));
  }
  (void)n_in; (void)out_size; (void)ws_size;
}